// DistMaps_37434934952233
// MI455X (gfx1250) — compile-verified
//
#include <hip/hip_runtime.h>
#include <hip/hip_bf16.h>

typedef __attribute__((ext_vector_type(2))) float v2f;
typedef __attribute__((ext_vector_type(8))) float v8f;

#define ROWS 512
#define COLS 512
#define P_PER_GROUP 24
#define P2 48

// tanh(2*sqrt(d2)) = 1 - 2/(exp2((4/ln2)*sqrt(d2)) + 1), fast HW ops (~1 ulp each)
__device__ __forceinline__ float finish(float d2) {
    float s = __builtin_amdgcn_sqrtf(d2);                       // v_sqrt_f32
    float e = __builtin_amdgcn_exp2f(5.770780163555853f * s);   // v_exp_f32
    return 1.0f - 2.0f * __builtin_amdgcn_rcpf(e + 1.0f);       // v_rcp_f32
}

__global__ __launch_bounds__(256) void distmaps_wmma_kernel(
    const float* __restrict__ coords,   // [B, 48, 3]
    float* __restrict__ out)            // [B, 2, 512, 512]
{
    const int lane = threadIdx.x & 31;
    const int wave = (blockIdx.x << 3) + (threadIdx.x >> 5);  // 8192 waves total
    const int bg = wave >> 9;                                 // 0..15
    const int r  = wave & (ROWS - 1);
    const int b  = bg >> 1;
    const int g  = bg & 1;

    const float inv25 = 1.0f / 25.0f;
    const float km2   = -2.0f / 25.0f;

    const int m = lane & 15;           // point-in-tile index / pixel-in-subtile index
    const bool hi = lane >= 16;        // upper half-wave: K=2,3 rows of A and B

    // ---- load two points per lane (WMMA set 0: points 0..15, set 1: points 16..31) ----
    float pr0, pc0, pr1, pc1;
    {
        const float* p0 = coords + ((size_t)b * P2 + (size_t)g * P_PER_GROUP + m) * 3;
        pr0 = p0[0];
        pc0 = p0[1];
        if (fmaxf(pr0, pc0) < 0.0f) { pr0 = -4096.0f; pc0 = -4096.0f; }  // invalid -> far fake
        const int i1 = 16 + m;
        if (i1 < P_PER_GROUP) {
            const float* p1 = coords + ((size_t)b * P2 + (size_t)g * P_PER_GROUP + i1) * 3;
            pr1 = p1[0];
            pc1 = p1[1];
            if (fmaxf(pr1, pc1) < 0.0f) { pr1 = -4096.0f; pc1 = -4096.0f; }
        } else {
            pr1 = -4096.0f; pc1 = -4096.0f;                               // pad -> far fake
        }
    }

    // ---- constant B matrices: 32-pixel tile split into two 16-pixel subtiles ----
    // B layout (4x16): v0 = K0 row (lanes 0-15) | K2 row (lanes 16-31)
    //                  v1 = K1 row (lanes 0-15) | K3 row (lanes 16-31)
    // tile-centered pixel coords: r' = 0, c' = m (subtile 1) / m+16 (subtile 2)
    v2f B1, B2;
    {
        const float c1 = (float)m, c2 = (float)(m + 16);
        B1.x = hi ? 1.0f : 0.0f;                      // K2 = 1        | K0 = r' = 0
        B1.y = hi ? c1 * c1 * inv25 : c1;             // K3 = c'^2/25  | K1 = c'
        B2.x = hi ? 1.0f : 0.0f;
        B2.y = hi ? c2 * c2 * inv25 : c2;
    }

    // ---- per-row point precompute ----
    const float rf = (float)r;
    const float a0 = pr0 - rf, a1 = pr1 - rf;       // pr' (row-centered)
    const float a0sq = a0 * a0, a1sq = a1 * a1;
    const float t00 = km2 * a0, t01 = km2 * a1;     // A[:,0] = -2*pr'/25

    float* orow = out + ((size_t)bg * ROWS + (size_t)r) * COLS;

    for (int c0 = 0; c0 < COLS; c0 += 32) {
        const float cf = (float)c0;
        const float bb0 = pc0 - cf;                      // pc' (tile-centered)
        const float bb1 = pc1 - cf;
        const float n0 = fmaf(bb0, bb0, a0sq) * inv25;   // (pr'^2+pc'^2)/25
        const float n1 = fmaf(bb1, bb1, a1sq) * inv25;

        // A layout (16x4): v0 = K0 (lanes 0-15) | K2 (lanes 16-31); v1 = K1 | K3
        v2f A0, A1;
        A0.x = hi ? n0 : t00;
        A0.y = hi ? 1.0f : km2 * bb0;
        A1.x = hi ? n1 : t01;
        A1.y = hi ? 1.0f : km2 * bb1;

        v8f cz = {};
        // D[point][pixel], 16 points x 16 pixels each; 4 products cover 32 pts x 32 px
        v8f dA0 = __builtin_amdgcn_wmma_f32_16x16x4_f32(false, A0, false, B1, (short)0, cz, false, false);
        v8f dA1 = __builtin_amdgcn_wmma_f32_16x16x4_f32(false, A1, false, B1, (short)0, cz, false, false);
        v8f dB0 = __builtin_amdgcn_wmma_f32_16x16x4_f32(false, A0, false, B2, (short)0, cz, false, false);
        v8f dB1 = __builtin_amdgcn_wmma_f32_16x16x4_f32(false, A1, false, B2, (short)0, cz, false, false);

        // per-subtile partial min over points (8 accum VGPRs x 2 WMMA sets)
        float mA, mB;
        {
            float e0 = fminf(dA0[0], dA1[0]), e1 = fminf(dA0[1], dA1[1]);
            float e2 = fminf(dA0[2], dA1[2]), e3 = fminf(dA0[3], dA1[3]);
            float e4 = fminf(dA0[4], dA1[4]), e5 = fminf(dA0[5], dA1[5]);
            float e6 = fminf(dA0[6], dA1[6]), e7 = fminf(dA0[7], dA1[7]);
            e0 = fminf(e0, e1); e2 = fminf(e2, e3); e4 = fminf(e4, e5); e6 = fminf(e6, e7);
            mA = fminf(fminf(e0, e2), fminf(e4, e6));
        }
        {
            float e0 = fminf(dB0[0], dB1[0]), e1 = fminf(dB0[1], dB1[1]);
            float e2 = fminf(dB0[2], dB1[2]), e3 = fminf(dB0[3], dB1[3]);
            float e4 = fminf(dB0[4], dB1[4]), e5 = fminf(dB0[5], dB1[5]);
            float e6 = fminf(dB0[6], dB1[6]), e7 = fminf(dB0[7], dB1[7]);
            e0 = fminf(e0, e1); e2 = fminf(e2, e3); e4 = fminf(e4, e5); e6 = fminf(e6, e7);
            mB = fminf(fminf(e0, e2), fminf(e4, e6));
        }

        // Cross-half combine serving BOTH subtiles with ONE swizzle:
        //   lanes 0-15 finalize subtile 1 (pixel c0+lane), lanes 16-31 subtile 2 (c0+lane).
        //   x holds "my subtile, my half" partial; y holds "my subtile, other half" after SWAPX16.
        float x = hi ? mB : mA;
        float y = hi ? mA : mB;
        y = __int_as_float(__builtin_amdgcn_ds_swizzle(__float_as_int(y), 0x401F)); // SWAPX16
        float mn = fminf(x, y);

        // fully uniform: every lane owns one distinct pixel
        orow[c0 + lane] = finish(mn);
    }
}

extern "C" void kernel_launch(void* const* d_in, const int* in_sizes, int n_in,
                              void* d_out, int out_size, void* d_ws, size_t ws_size,
                              hipStream_t stream) {
    // d_in[0] = x (only its shape matters; never read), d_in[1] = coords [8,48,3] f32
    const float* coords = (const float*)d_in[1];
    float* out = (float*)d_out;   // [8,2,512,512] f32

    // 16 (b,g) pairs * 512 rows = 8192 waves; 8 waves (256 threads) per block
    dim3 grid(1024), block(256);
    distmaps_wmma_kernel<<<grid, block, 0, stream>>>(coords, out);
}